// SparseGlobalAvgPool2d_50895362457607
// MI455X (gfx1250) — compile-verified
//
#include <hip/hip_runtime.h>
#include <hip/hip_bf16.h>
#include <stdint.h>

// Column-sum of [N, 64] fp32 via WMMA ones-matmul accumulation.
//   out[c] = sum_n feat[n][c] / (H*W)
//
// V_WMMA_F32_16X16X4_F32: D(16x16,f32) = A(16x4,f32) x B(4x16,f32) + C.
// A = all ones  =>  every row of D accumulates the column sums of the
// 4x16 B tiles streamed through it. Because we sum over all K with a
// constant-ones A, the exact K<->lane assignment of the B layout cannot
// change the result; D and B share the N = lane%16 column mapping.
//
// Bandwidth-bound: 128 MB in / 23.3 TB/s => ~5.5 us floor. The inner loop
// is built to be branch-free (no EXEC-masked loads) with 16 independent
// dword loads in flight per wave per iteration.

typedef float v2f __attribute__((ext_vector_type(2)));
typedef float v8f __attribute__((ext_vector_type(8)));

#define CH        64      // channels (fixed by the reference)
#define BLK       128     // 4 waves per block; wave w owns cols [16w,16w+16)
#define NPARTIAL  1024    // max partial blocks (256 KB of workspace)
#define GROUP     32      // rows consumed per block per main-loop iteration

__global__ void __launch_bounds__(BLK)
colsum_partial_wmma(const float* __restrict__ feat,
                    float* __restrict__ partial,   // [gridDim.x][64]
                    int nrows) {
    const int lane = threadIdx.x & 31;
    const int wave = threadIdx.x >> 5;       // 0..3  -> column tile
    const int n0   = wave * 16;              // first column of this wave's tile
    const int half = lane >> 4;              // 0: lanes 0-15, 1: lanes 16-31
    const int nc   = lane & 15;              // column within tile (N = lane%16)

    v8f acc = {0.f, 0.f, 0.f, 0.f, 0.f, 0.f, 0.f, 0.f};
    const v2f ones = {1.0f, 1.0f};

    // ---- Main loop: full 32-row groups, fully unconditional loads ----
    const int64_t ngroups = (int64_t)nrows / GROUP;
    const int64_t gstep   = gridDim.x;
    // Per-lane base: row (g*32 + half*2), column (n0 + nc).
    const float* p = feat + ((int64_t)blockIdx.x * GROUP + half * 2) * CH
                          + n0 + nc;
    const int64_t pstep = gstep * GROUP * CH;   // floats per grid-stride hop

    for (int64_t g = blockIdx.x; g < ngroups; g += gstep, p += pstep) {
#pragma unroll
        for (int u = 0; u < GROUP / 4; ++u) {
            // 4-row subgroup u: this lane supplies rows (4u + 2*half) and +1,
            // at immediate byte offsets 1024*u and 1024*u + 256.
            v2f b;
            b.x = p[u * 4 * CH];
            b.y = p[u * 4 * CH + CH];
            // 8 args: (neg_a, A, neg_b, B, c_mod, C, reuse_a, reuse_b)
            acc = __builtin_amdgcn_wmma_f32_16x16x4_f32(
                false, ones, false, b, (short)0, acc, false, false);
        }
    }

    // ---- Tail: rows [ngroups*32, nrows), < 32 rows, block 0 only ----
    // Branch-free: clamp addresses in-bounds, zero OOB values via select
    // (v_cndmask). EXEC stays all-1s; all control flow here is uniform.
    const int64_t tbase = ngroups * GROUP;
    if ((tbase < nrows) && (blockIdx.x == 0)) {
        const float* tp = feat + n0 + nc;
        const int64_t last = (int64_t)nrows - 1;
#pragma unroll
        for (int u = 0; u < GROUP / 4; ++u) {
            const int64_t r0 = tbase + u * 4 + half * 2;
            const int64_t r1 = r0 + 1;
            const int64_t c0 = (r0 < nrows) ? r0 : last;
            const int64_t c1 = (r1 < nrows) ? r1 : last;
            const float v0 = tp[c0 * CH];
            const float v1 = tp[c1 * CH];
            v2f b;
            b.x = (r0 < nrows) ? v0 : 0.0f;
            b.y = (r1 < nrows) ? v1 : 0.0f;
            acc = __builtin_amdgcn_wmma_f32_16x16x4_f32(
                false, ones, false, b, (short)0, acc, false, false);
        }
    }

    // Every row of D holds the same column sums; VGPR0 / lanes 0-15 is
    // (M=0, N=lane): the 16 column sums for this wave's tile.
    if (lane < 16) {
        partial[(size_t)blockIdx.x * CH + n0 + nc] = acc[0];
    }
}

__global__ void __launch_bounds__(CH)
colsum_final(const float* __restrict__ partial,  // [nblocks][64]
             const int* __restrict__ Hp,
             const int* __restrict__ Wp,
             float* __restrict__ out,
             int nblocks) {
    const int c = threadIdx.x;               // 0..63
    float s = 0.0f;
    for (int b = 0; b < nblocks; ++b) {      // fixed order -> deterministic
        s += partial[(size_t)b * CH + c];
    }
    const float scale = 1.0f / ((float)(*Hp) * (float)(*Wp));
    out[c] = s * scale;
}

extern "C" void kernel_launch(void* const* d_in, const int* in_sizes, int n_in,
                              void* d_out, int out_size, void* d_ws, size_t ws_size,
                              hipStream_t stream) {
    const float* feat = (const float*)d_in[0];   // [N, 64] fp32
    const int*   Hp   = (const int*)d_in[1];     // scalar H
    const int*   Wp   = (const int*)d_in[2];     // scalar W

    const int nrows = in_sizes[0] / CH;

    int nblocks = NPARTIAL;
    const size_t per_block = (size_t)CH * sizeof(float);
    if (ws_size < (size_t)nblocks * per_block) {
        nblocks = (int)(ws_size / per_block);
        if (nblocks < 1) nblocks = 1;
    }

    float* partial = (float*)d_ws;

    colsum_partial_wmma<<<nblocks, BLK, 0, stream>>>(feat, partial, nrows);
    colsum_final<<<1, CH, 0, stream>>>(partial, Hp, Wp, (float*)d_out, nblocks);
}